// DropConnect_57690000720381
// MI455X (gfx1250) — compile-verified
//
#include <hip/hip_runtime.h>
#include <cstdint>

// DropConnect: y[b,n] = sum_k x[b,k]*(u_w[b,k,n]<=0.5)*W[k,n] + (u_b[b,n]<=0.5)*bias[n]
// HBM-bound (512MB u_w stream, ~23us floor @ 23.3TB/s). Per-element masks preclude WMMA.
// CDNA5 data movement: u_w streamed via per-lane GLOBAL_LOAD_ASYNC_TO_LDS_B128 (ASYNCcnt)
// with th:TH_LOAD_NT so the read-once stream does not evict the L2-resident weight array;
// weight tiles fetched by the Tensor Data Mover (TENSOR_LOAD_TO_LDS, TENSORcnt); both
// double-buffered in LDS ahead of a cmp/cndmask/fmac VALU loop.

constexpr int BS  = 32;
constexpr int D   = 2048;
constexpr int TN  = 128;           // n-columns per workgroup
constexpr int KC  = 32;            // k-rows per chunk
constexpr int NCH = D / KC;        // 64 chunks
constexpr float PKEEP = 0.5f;

typedef uint32_t u32x4 __attribute__((ext_vector_type(4)));
typedef uint32_t u32x8 __attribute__((ext_vector_type(8)));

__device__ __forceinline__ void async_ld16_nt(uint32_t lds_addr, const float* gaddr) {
  // VDST = LDS byte address (low 32 bits of flat shared address), VADDR = 64-bit global.
  // NT: read-once stream, keep out of the way of the weight working set in L2.
  asm volatile("global_load_async_to_lds_b128 %0, %1, off th:TH_LOAD_NT"
               :: "v"(lds_addr), "v"(gaddr) : "memory");
}
__device__ __forceinline__ void wait_async_le4() {
  asm volatile("s_wait_asynccnt 4" ::: "memory");
}
__device__ __forceinline__ void wait_async_le0() {
  asm volatile("s_wait_asynccnt 0" ::: "memory");
}

#if defined(__has_builtin) && __has_builtin(__builtin_amdgcn_s_wait_tensorcnt)
#define WAIT_TENSOR(n) __builtin_amdgcn_s_wait_tensorcnt(n)
#else
#define WAIT_TENSOR(n) asm volatile("s_wait_tensorcnt %0" :: "i"(n) : "memory")
#endif

// One 2D TDM descriptor: KC x TN tile of f32, tensor row stride D elements.
__device__ __forceinline__ void tdm_load_tile(uint32_t lds_addr, const float* gaddr) {
  const uint64_t ga = (uint64_t)(uintptr_t)gaddr;
  u32x4 g0;
  g0[0] = 0x1u;                                          // count=1 valid descriptor
  g0[1] = lds_addr;                                      // lds_addr (bytes)
  g0[2] = (uint32_t)ga;                                  // global_addr[31:0]
  g0[3] = (uint32_t)((ga >> 32) & 0x01FFFFFFu)           // global_addr[56:32]
          | (2u << 30);                                  // type=2 ("image")
  constexpr uint32_t TD0 = 2048u;                        // tensor_dim0 (elements)
  constexpr uint32_t TD1 = 1u << 20;                     // tensor_dim1 (rows, generous)
  u32x8 g1;
  g1[0] = (2u << 16);                                    // data_size=2 -> 4 bytes
  g1[1] = (TD0 & 0xFFFFu) << 16;                         // tensor_dim0[15:0] @ bits63:48
  g1[2] = (TD0 >> 16) | ((TD1 & 0xFFFFu) << 16);         // dim0 hi | dim1 lo
  g1[3] = (TD1 >> 16) | ((uint32_t)TN << 16);            // dim1 hi | tile_dim0=128
  g1[4] = (uint32_t)KC;                                  // tile_dim1=32, tile_dim2=0
  g1[5] = (uint32_t)D;                                   // tensor_dim0_stride lo32 = 2048
  g1[6] = 0u;                                            // stride0 hi | stride1 lo
  g1[7] = 0u;
  // Groups 2/3 NULL: 2D tensor form.
  asm volatile("tensor_load_to_lds %0, %1" :: "s"(g0), "s"(g1) : "memory");
}

__global__ __launch_bounds__(256, 4) void dropconnect_kernel(
    const float* __restrict__ x,      // (BS, D)
    const float* __restrict__ weight, // (D, D)  [k, n]
    const float* __restrict__ bias,   // (D)
    const float* __restrict__ u_w,    // (BS, D, D)
    const float* __restrict__ u_b,    // (BS, D)
    float* __restrict__ out) {        // (BS, D)
  __shared__ __align__(16) float uwBuf[2][KC * TN];   // 2 x 16 KB
  __shared__ __align__(16) float wtBuf[2][KC * TN];   // 2 x 16 KB
  __shared__ __align__(16) float xls[D];              // 8 KB
  __shared__ __align__(16) float red[256 * 4];        // 4 KB

  const int tid = threadIdx.x;
  const int b   = blockIdx.y;
  const int n0  = blockIdx.x * TN;

  // Stage x[b,:] into LDS once.
  for (int i = tid; i < D; i += 256) xls[i] = x[(size_t)b * D + i];

  const float* uw_base = u_w + ((size_t)b * D) * D + n0;   // row k, col n0
  const float* wt_base = weight + n0;

  const uint32_t uwL[2] = { (uint32_t)(uintptr_t)&uwBuf[0][0],
                            (uint32_t)(uintptr_t)&uwBuf[1][0] };
  const uint32_t wtL[2] = { (uint32_t)(uintptr_t)&wtBuf[0][0],
                            (uint32_t)(uintptr_t)&wtBuf[1][0] };

  const int krow = tid >> 5;         // 0..7   (k sub-row per issue round)
  const int nl   = (tid & 31) << 2;  // 0..124 (4 consecutive columns, 16B)

  // SGPR-uniform wave-0 predicate: guarantees a scalar branch so only wave 0
  // reaches the TDM issue (tensor ops ignore EXEC; a masked fallthrough would
  // duplicate the DMA 8x).
  const bool wave0 = (__builtin_amdgcn_readfirstlane(tid) < 32);

  auto issue_uw = [&](int c, int bufsel) {      // 4 async ops per wave per chunk
    const float* ug = uw_base + (size_t)(c * KC) * D;
#pragma unroll
    for (int i = 0; i < 4; ++i) {
      const int k = krow + i * 8;               // covers k = 0..31
      async_ld16_nt(uwL[bufsel] + (uint32_t)((k * TN + nl) * 4),
                    ug + (size_t)k * D + nl);
    }
  };
  auto issue_wt = [&](int c, int bufsel) {      // 1 TDM op per chunk (wave 0 only)
    tdm_load_tile(wtL[bufsel], wt_base + (size_t)(c * KC) * D);
  };

  // Prologue: fill both buffers.
  issue_uw(0, 0);
  issue_uw(1, 1);
  if (wave0) { issue_wt(0, 0); issue_wt(1, 1); }

  float4 acc = make_float4(0.f, 0.f, 0.f, 0.f);

  for (int c = 0; c < NCH; ++c) {
    // In-order completion per counter class: oldest 4 async ops / oldest TDM op
    // belong to chunk c.
    if (c + 1 < NCH) { wait_async_le4(); WAIT_TENSOR(1); }
    else             { wait_async_le0(); WAIT_TENSOR(0); }
    __syncthreads();  // all producers' portions visible in LDS

    const float* uwp = uwBuf[c & 1];
    const float* wtp = wtBuf[c & 1];
#pragma unroll
    for (int kk = 0; kk < 4; ++kk) {
      const int k = krow * 4 + kk;                 // this thread's 4 k-rows
      const float4 u = *(const float4*)(uwp + k * TN + nl);
      const float4 w = *(const float4*)(wtp + k * TN + nl);
      const float xv = xls[c * KC + k];            // wave-uniform broadcast
      acc.x += ((u.x <= PKEEP) ? w.x : 0.f) * xv;
      acc.y += ((u.y <= PKEEP) ? w.y : 0.f) * xv;
      acc.z += ((u.z <= PKEEP) ? w.z : 0.f) * xv;
      acc.w += ((u.w <= PKEEP) ? w.w : 0.f) * xv;
    }
    __syncthreads();  // everyone done reading buf (c&1) before overwrite
    if (c + 2 < NCH) {
      issue_uw(c + 2, c & 1);
      if (wave0) issue_wt(c + 2, c & 1);
    }
  }

  // Reduce the 8 k-group partials per column through LDS.
  *(float4*)(red + tid * 4) = acc;
  __syncthreads();
  if (tid < 32) {
    float4 tot = make_float4(0.f, 0.f, 0.f, 0.f);
#pragma unroll
    for (int g = 0; g < 8; ++g) {
      const float4 p = *(const float4*)(red + (g * 32 + tid) * 4);
      tot.x += p.x; tot.y += p.y; tot.z += p.z; tot.w += p.w;
    }
    const int n = n0 + tid * 4;
    const float4 ub = *(const float4*)(u_b + (size_t)b * D + n);
    const float4 bv = *(const float4*)(bias + n);
    float4 y;
    y.x = tot.x + ((ub.x <= PKEEP) ? bv.x : 0.f);
    y.y = tot.y + ((ub.y <= PKEEP) ? bv.y : 0.f);
    y.z = tot.z + ((ub.z <= PKEEP) ? bv.z : 0.f);
    y.w = tot.w + ((ub.w <= PKEEP) ? bv.w : 0.f);
    *(float4*)(out + (size_t)b * D + n) = y;
  }
}

extern "C" void kernel_launch(void* const* d_in, const int* in_sizes, int n_in,
                              void* d_out, int out_size, void* d_ws, size_t ws_size,
                              hipStream_t stream) {
  (void)in_sizes; (void)n_in; (void)out_size; (void)d_ws; (void)ws_size;
  const float* x      = (const float*)d_in[0];
  const float* weight = (const float*)d_in[1];
  const float* bias   = (const float*)d_in[2];
  const float* u_w    = (const float*)d_in[3];
  const float* u_b    = (const float*)d_in[4];
  float* out          = (float*)d_out;

  dim3 grid(D / TN, BS);   // (16, 32) = 512 workgroups
  dim3 block(256);
  dropconnect_kernel<<<grid, block, 0, stream>>>(x, weight, bias, u_w, u_b, out);
}